// DGCNN_57277683859500
// MI455X (gfx1250) — compile-verified
//
#include <hip/hip_runtime.h>
#include <math.h>

// ---------------- fixed problem sizes ----------------
#define GG    128
#define NN    512
#define EE    512
#define MEMB  16
#define GN_   (GG * NN)        // 65536
#define GE_   (GG * EE)        // 65536
#define NNZ_  (GE_ * MEMB)     // 1048576
#define NNZG  (EE * MEMB)      // 8192 nnz per graph
#define F_IN  128
#define TOT   97
#define KTOP  30
#define C1_   16
#define C2_   32
#define KW2_  5
#define DENSE 352              // ((30-2)/2+1-5+1)*32
#define ODIM  128

typedef float v2f  __attribute__((ext_vector_type(2)));
typedef float v8f  __attribute__((ext_vector_type(8)));

// ---- WMMA f32 16x16x4 tile: C += A[16,Kd] x B(=W^T)[Kd,16], K-swept ----
// arow: row (lane&15) of A tile, wrow: col (lane&15) of B tile (= row of W).
// kOff = (lane>>4)*2 per the ISA 16x4 f32 operand layout.
__device__ __forceinline__ v8f wmma_tile_f32(const float* __restrict__ arow,
                                             const float* __restrict__ wrow,
                                             int Kd, int kOff) {
    v8f c = {0.f, 0.f, 0.f, 0.f, 0.f, 0.f, 0.f, 0.f};
    for (int k0 = 0; k0 < Kd; k0 += 4) {
        v2f a, b;
        a.x = arow[k0 + kOff];
        a.y = arow[k0 + kOff + 1];
        b.x = wrow[k0 + kOff];
        b.y = wrow[k0 + kOff + 1];
        c = __builtin_amdgcn_wmma_f32_16x16x4_f32(false, a, false, b,
                                                  (short)0, c, false, false);
    }
    return c;
}

// ---------------- node degrees: per-graph LDS histogram ----------------
__global__ void degree_graph_kernel(const int* __restrict__ node_idx,
                                    float* __restrict__ degs) {
    __shared__ int cnt[NN];
    int g = blockIdx.x, t = threadIdx.x;            // 512 threads
    cnt[t] = 0;
    __syncthreads();
    const int base = g * NNZG;
    for (int i = t; i < NNZG; i += NN)
        atomicAdd(&cnt[node_idx[base + i] - g * NN], 1);
    __syncthreads();
    degs[(size_t)g * NN + t] = (float)cnt[t];
}

// n2e gather: dst[e*f + t] = sum_{j<16} src[node_idx[e*16+j]*stride + coloff + t]
__global__ void gather_sum_kernel(const float* __restrict__ src, int stride, int coloff,
                                  const int* __restrict__ node_idx,
                                  float* __restrict__ dst, int f, int total) {
    int id = blockIdx.x * blockDim.x + threadIdx.x;
    if (id >= total) return;
    int e = id / f;
    int t = id - e * f;
    const int* ni = node_idx + e * MEMB;
    float s = 0.0f;
#pragma unroll
    for (int j = 0; j < MEMB; ++j)
        s += src[(size_t)ni[j] * stride + coloff + t];
    dst[id] = s;
}

// ---------------- n2e GEMM: emsg = tanh(A[M,Kd] @ W[32,Kd]^T + b) ----------------
// One 16x16 C tile per wave; 4 waves per 128-thread block.
__global__ void n2e_gemm_kernel(const float* __restrict__ A, int Kd,
                                const float* __restrict__ W,
                                const float* __restrict__ bias,
                                float* __restrict__ out, int M) {
    const int lane = threadIdx.x & 31;
    const int wave = threadIdx.x >> 5;
    int tile = blockIdx.x * (blockDim.x >> 5) + wave;
    int numTiles = (M >> 4) * 2;                    // Nd = 32
    if (tile >= numTiles) return;                   // wave-uniform
    int tm = tile >> 1;
    int tn = tile & 1;

    const int kOff = (lane >> 4) * 2;
    const float* arow = A + (size_t)(tm * 16 + (lane & 15)) * Kd;
    const float* wrow = W + (size_t)(tn * 16 + (lane & 15)) * Kd;
    v8f c = wmma_tile_f32(arow, wrow, Kd, kOff);

    int n = tn * 16 + (lane & 15);
    float bcol = bias[n];
    int mbase = tm * 16 + ((lane >> 4) << 3);       // VGPR v: M = v + 8*(lane>=16)
#pragma unroll
    for (int v = 0; v < 8; ++v)
        out[(size_t)(mbase + v) * 32 + n] = tanhf(c[v] + bcol);
}

// ---------------- fused e2n: LDS scatter-add + WMMA-from-LDS + epilogue ----------------
// One 512-thread block (16 waves) per graph. acc = 512 nodes x 32 feats = 64 KB LDS.
__global__ void e2n_fused_kernel(const float* __restrict__ emsg,   // [GE,32]
                                 const int* __restrict__ node_idx,
                                 const float* __restrict__ We,     // [32,32]
                                 const float* __restrict__ be,
                                 const float* __restrict__ degs,
                                 float* __restrict__ feat, int coloff) {
    __shared__ float acc[NN * 32];                  // 65536 B
    int g = blockIdx.x, t = threadIdx.x;            // 512 threads
    int lane = t & 31, wave = t >> 5;

    for (int i = t; i < NN * 32; i += 512) acc[i] = 0.0f;
    __syncthreads();

    // scatter: wave w handles nnz (i0 + w); lane = feature -> conflict-free banks,
    // node_idx load is wave-uniform (scalarized), emsg read is lane-coalesced.
    const int base = g * NNZG;
    for (int i0 = 0; i0 < NNZG; i0 += 16) {
        int i = base + i0 + wave;
        int n = node_idx[i] - g * NN;               // local node
        int e = i >> 4;                             // global edge (edge_idx[i])
        atomicAdd(&acc[n * 32 + lane], emsg[(size_t)e * 32 + lane]);
    }
    __syncthreads();

    // WMMA: [512,32] (LDS) @ We^T[32,32] -> 32x2 = 64 tiles over 16 waves
    const int kOff = (lane >> 4) * 2;
    for (int tt = wave; tt < 64; tt += 16) {        // wave-uniform trip count
        int tm = tt >> 1, tn = tt & 1;
        const float* arow = &acc[(tm * 16 + (lane & 15)) * 32];
        const float* wrow = We + (size_t)(tn * 16 + (lane & 15)) * 32;
        v8f c = wmma_tile_f32(arow, wrow, 32, kOff);

        int n = tn * 16 + (lane & 15);
        float bcol = be[n];
        int mbase = tm * 16 + ((lane >> 4) << 3);
#pragma unroll
        for (int v = 0; v < 8; ++v) {
            int row = g * NN + mbase + v;
            float s = 1.0f / fmaxf(degs[row], 1.0f);
            feat[(size_t)row * TOT + coloff + n] = tanhf((c[v] + bcol) * s);
        }
    }
}

// ---------------- last layer (latent dim 1) ----------------
__global__ void n2e_last_kernel(const float* __restrict__ n2e,  // [GE,32]
                                const float* __restrict__ Wn3,  // [1,32]
                                const float* __restrict__ bn3,
                                float* __restrict__ msg, int ge) {
    int e = blockIdx.x * blockDim.x + threadIdx.x;
    if (e >= ge) return;
    float s = bn3[0];
#pragma unroll
    for (int k = 0; k < 32; ++k) s += n2e[(size_t)e * 32 + k] * Wn3[k];
    msg[e] = tanhf(s);
}

__global__ void e2n_last_graph_kernel(const float* __restrict__ emsg,  // [GE]
                                      const int* __restrict__ node_idx,
                                      const float* __restrict__ We3,   // [1,1]
                                      const float* __restrict__ be3,
                                      const float* __restrict__ degs,
                                      float* __restrict__ feat) {
    __shared__ float acc1[NN];
    int g = blockIdx.x, t = threadIdx.x;            // 512 threads
    acc1[t] = 0.0f;
    __syncthreads();
    const int base = g * NNZG;
    for (int i = t; i < NNZG; i += NN) {
        int gi = base + i;
        atomicAdd(&acc1[node_idx[gi] - g * NN], emsg[gi >> 4]);
    }
    __syncthreads();
    int n = g * NN + t;
    float d = fmaxf(degs[n], 1.0f);
    feat[(size_t)n * TOT + (TOT - 1)] = tanhf((acc1[t] * We3[0] + be3[0]) / d);
}

// ---------------- sort pooling: per-graph top-30 of feat[:, :, 96] ----------------
__global__ void topk_kernel(const float* __restrict__ feat, int* __restrict__ sel) {
    __shared__ float sval[NN];
    __shared__ float rv[NN];
    __shared__ int   ri[NN];
    int g = blockIdx.x, t = threadIdx.x;
    sval[t] = feat[((size_t)g * NN + t) * TOT + (TOT - 1)];
    __syncthreads();
    for (int k = 0; k < KTOP; ++k) {
        rv[t] = sval[t];
        ri[t] = t;
        __syncthreads();
        for (int s = NN / 2; s > 0; s >>= 1) {
            if (t < s) {
                float v2 = rv[t + s]; int i2 = ri[t + s];
                if (v2 > rv[t] || (v2 == rv[t] && i2 < ri[t])) { rv[t] = v2; ri[t] = i2; }
            }
            __syncthreads();
        }
        if (t == 0) { sel[g * KTOP + k] = ri[0]; sval[ri[0]] = -__builtin_inff(); }
        __syncthreads();
    }
}

// ---------------- fused head: conv1 -> relu -> maxpool -> conv2 -> relu -> dense -> relu ----------------
__global__ void head_kernel(const float* __restrict__ feat, const int* __restrict__ sel,
                            const float* __restrict__ Wc1, const float* __restrict__ bc1,
                            const float* __restrict__ Wc2, const float* __restrict__ bc2,
                            const float* __restrict__ Wout, const float* __restrict__ bout,
                            float* __restrict__ out) {
    __shared__ float pooled[KTOP * TOT];   // 2910
    __shared__ float y1[C1_ * KTOP];       // 480
    __shared__ float y2[C1_ * (KTOP / 2)]; // 240
    __shared__ float z[DENSE];             // 352
    int g = blockIdx.x, t = threadIdx.x;   // 128 threads

    for (int i = t; i < KTOP * TOT; i += ODIM) {
        int k = i / TOT, c = i - k * TOT;
        int node = sel[g * KTOP + k];
        pooled[i] = feat[((size_t)g * NN + node) * TOT + c];
    }
    __syncthreads();

    for (int i = t; i < C1_ * KTOP; i += ODIM) {
        int o = i / KTOP, k = i - o * KTOP;
        float s = bc1[o];
        for (int c = 0; c < TOT; ++c) s += pooled[k * TOT + c] * Wc1[o * TOT + c];
        y1[o * KTOP + k] = fmaxf(s, 0.0f);
    }
    __syncthreads();

    for (int i = t; i < C1_ * (KTOP / 2); i += ODIM) {
        int o = i / (KTOP / 2), p = i - o * (KTOP / 2);
        y2[i] = fmaxf(y1[o * KTOP + 2 * p], y1[o * KTOP + 2 * p + 1]);
    }
    __syncthreads();

    for (int i = t; i < DENSE; i += ODIM) {       // 32 out-ch x 11 positions
        int o2 = i / 11, tt = i - o2 * 11;
        float s = bc2[o2];
        for (int c1 = 0; c1 < C1_; ++c1)
#pragma unroll
            for (int dk = 0; dk < KW2_; ++dk)
                s += y2[c1 * (KTOP / 2) + tt + dk] * Wc2[(o2 * C1_ + c1) * KW2_ + dk];
        z[i] = fmaxf(s, 0.0f);
    }
    __syncthreads();

    {   // dense 352 -> 128 (double relu == relu)
        float s = bout[t];
        for (int d = 0; d < DENSE; ++d) s += z[d] * Wout[t * DENSE + d];
        out[(size_t)g * ODIM + t] = fmaxf(s, 0.0f);
    }
}

// ---------------- launcher ----------------
extern "C" void kernel_launch(void* const* d_in, const int* in_sizes, int n_in,
                              void* d_out, int out_size, void* d_ws, size_t ws_size,
                              hipStream_t stream) {
    const float* node_feat = (const float*)d_in[0];
    const int*   node_idx  = (const int*)d_in[1];
    // d_in[2] (edge_idx) is repeat(arange(GE),16) by construction -> derived in-kernel.
    const float* Wn[4] = {(const float*)d_in[3], (const float*)d_in[5],
                          (const float*)d_in[7], (const float*)d_in[9]};
    const float* bn[4] = {(const float*)d_in[4], (const float*)d_in[6],
                          (const float*)d_in[8], (const float*)d_in[10]};
    const float* We[4] = {(const float*)d_in[11], (const float*)d_in[13],
                          (const float*)d_in[15], (const float*)d_in[17]};
    const float* be[4] = {(const float*)d_in[12], (const float*)d_in[14],
                          (const float*)d_in[16], (const float*)d_in[18]};
    const float* Wc1  = (const float*)d_in[19];
    const float* bc1  = (const float*)d_in[20];
    const float* Wc2  = (const float*)d_in[21];
    const float* bc2  = (const float*)d_in[22];
    const float* Wout = (const float*)d_in[23];
    const float* bout = (const float*)d_in[24];
    float* out = (float*)d_out;

    float* ws = (float*)d_ws;
    size_t off = 0;
    float* feat = ws + off; off += (size_t)GN_ * TOT;   // [GN,97]
    float* degs = ws + off; off += (size_t)GN_;         // [GN]
    float* n2e  = ws + off; off += (size_t)GE_ * F_IN;  // [GE,<=128]
    float* emsg = ws + off; off += (size_t)GE_ * 32;    // [GE,<=32]
    int*   sel  = (int*)(ws + off);                     // [G,30]

    const int TPB = 256;
    auto nb = [](int total, int tpb) { return (total + tpb - 1) / tpb; };

    // node degrees (per-graph LDS histogram, deterministic int atomics)
    degree_graph_kernel<<<GG, NN, 0, stream>>>(node_idx, degs);

    // layers 0..2 (latent 32)
    for (int l = 0; l < 3; ++l) {
        const float* src = (l == 0) ? node_feat : feat;
        int stride = (l == 0) ? F_IN : TOT;
        int coloff = (l == 0) ? 0 : 32 * (l - 1);
        int f      = (l == 0) ? F_IN : 32;

        int tg = GE_ * f;
        gather_sum_kernel<<<nb(tg, TPB), TPB, 0, stream>>>(src, stride, coloff,
                                                           node_idx, n2e, f, tg);
        {   // edge_msg = tanh(n2e @ Wn^T + bn)   [GE,f]x[f,32]
            int tiles = (GE_ / 16) * 2;
            n2e_gemm_kernel<<<nb(tiles, 4), 128, 0, stream>>>(n2e, f, Wn[l], bn[l],
                                                              emsg, GE_);
        }
        // fused: LDS scatter + WMMA-from-LDS + tanh((.+b)/deg) -> feat cols 32l..
        e2n_fused_kernel<<<GG, NN, 0, stream>>>(emsg, node_idx, We[l], be[l],
                                                degs, feat, 32 * l);
    }

    // layer 3 (latent 1)
    {
        int tg = GE_ * 32;
        gather_sum_kernel<<<nb(tg, TPB), TPB, 0, stream>>>(feat, TOT, 64,
                                                           node_idx, n2e, 32, tg);
        n2e_last_kernel<<<nb(GE_, TPB), TPB, 0, stream>>>(n2e, Wn[3], bn[3], emsg, GE_);
        e2n_last_graph_kernel<<<GG, NN, 0, stream>>>(emsg, node_idx, We[3], be[3],
                                                     degs, feat);
    }

    // sort pooling + head
    topk_kernel<<<GG, NN, 0, stream>>>(feat, sel);
    head_kernel<<<GG, ODIM, 0, stream>>>(feat, sel, Wc1, bc1, Wc2, bc2, Wout, bout, out);
}